// Heatmaps_57363583205469
// MI455X (gfx1250) — compile-verified
//
#include <hip/hip_runtime.h>
#include <math.h>

// Problem shape (fixed by the reference): [16,17,512,512] f32, kernel_size=3.
#define H 512
#define W 512
#define TILE_ROWS 16
#define IN_ROWS   18            // TILE_ROWS + 2 halo rows
#define LDS_STRIDE 516          // 512 + 4 floats pad -> 2064B row stride (16B aligned, bank-skewed)

typedef float float4v __attribute__((ext_vector_type(4)));
typedef int   v4i     __attribute__((vector_size(4 * sizeof(int))));

typedef __attribute__((address_space(1))) v4i* gptr_v4i;
typedef __attribute__((address_space(3))) v4i* lptr_v4i;

__device__ __forceinline__ float fmax3(float a, float b, float c) {
    return fmaxf(fmaxf(a, b), c);
}

// Horizontal 3-max over a 4-wide chunk starting at x4 of a 512-float LDS row.
// Edge replication (clamp) is exactly equivalent to -inf padding for max.
__device__ __forceinline__ float4v hrow(const float* __restrict__ rowp, int x4) {
    float4v v = *(const float4v*)(rowp + x4);
    float l = rowp[(x4 == 0) ? 0 : (x4 - 1)];
    float r = rowp[(x4 == W - 4) ? (W - 1) : (x4 + 4)];
    float4v h;
    h.x = fmax3(l,   v.x, v.y);
    h.y = fmax3(v.x, v.y, v.z);
    h.z = fmax3(v.y, v.z, v.w);
    h.w = fmax3(v.z, v.w, r);
    return h;
}

__device__ __forceinline__ float sigmoid_fast(float x) {
    return 1.0f / (1.0f + __expf(-x));
}

__global__ __launch_bounds__(256)
void heatmap_peak_kernel(const float* __restrict__ in, float* __restrict__ out) {
    __shared__ float tile[IN_ROWS * LDS_STRIDE];

    const int block = blockIdx.x;
    const int tileY = block & 31;           // 32 row-tiles per 512-row plane
    const int plane = block >> 5;           // B*K plane index
    const int t     = threadIdx.x;

    const long long planeBase = (long long)plane * (long long)(H * W);
    const int row0 = tileY * TILE_ROWS;     // first output row of this tile

    // ---- Stage 18 input rows (row0-1 .. row0+16, clamped) into LDS ----
    const int xq   = t & 127;               // which float4 column chunk
    const int x4   = xq << 2;               // 0..508, 16B aligned
    const int rsel = t >> 7;                // 0 or 1: which of 2 rows per round

#pragma unroll
    for (int rr = 0; rr < 9; ++rr) {
        const int lrow = rr * 2 + rsel;     // 0..17
        int gy = row0 - 1 + lrow;
        gy = gy < 0 ? 0 : (gy > H - 1 ? H - 1 : gy);
        const float* gsrc = in + planeBase + (long long)gy * W + x4;
        float* ldst = &tile[lrow * LDS_STRIDE + x4];
#if __has_builtin(__builtin_amdgcn_global_load_async_to_lds_b128)
        // gfx1250 async global->LDS DMA path (ASYNCcnt-tracked), 16B per lane.
        __builtin_amdgcn_global_load_async_to_lds_b128(
            (gptr_v4i)gsrc, (lptr_v4i)ldst, /*offset=*/0, /*cpol=*/0);
#else
#pragma message("async-to-LDS builtin not found; using VGPR staging fallback")
        *(float4v*)ldst = *(const float4v*)gsrc;
#endif
    }
#if __has_builtin(__builtin_amdgcn_global_load_async_to_lds_b128)
#if __has_builtin(__builtin_amdgcn_s_wait_asynccnt)
    __builtin_amdgcn_s_wait_asynccnt(0);
#else
    asm volatile("s_wait_asynccnt 0" ::: "memory");
#endif
#endif
    __syncthreads();

    // ---- Compute: each thread owns a 4-wide chunk x 8 output rows ----
    const int base = rsel * 8;              // first LDS row of this thread's window stack
    float4v h0 = hrow(&tile[(base + 0) * LDS_STRIDE], x4);
    float4v h1 = hrow(&tile[(base + 1) * LDS_STRIDE], x4);

#pragma unroll
    for (int k = 0; k < 8; ++k) {
        const int crow = base + k + 1;      // LDS row holding the center row
        float4v h2 = hrow(&tile[(crow + 1) * LDS_STRIDE], x4);
        float4v c  = *(const float4v*)&tile[crow * LDS_STRIDE + x4];

        float4v m;
        m.x = fmax3(h0.x, h1.x, h2.x);
        m.y = fmax3(h0.y, h1.y, h2.y);
        m.z = fmax3(h0.z, h1.z, h2.z);
        m.w = fmax3(h0.w, h1.w, h2.w);

        float4v s;
        s.x = sigmoid_fast(c.x);
        s.y = sigmoid_fast(c.y);
        s.z = sigmoid_fast(c.z);
        s.w = sigmoid_fast(c.w);

        float4v o;
        o.x = (m.x == c.x && s.x > 0.05f) ? s.x : 0.0f;
        o.y = (m.y == c.y && s.y > 0.05f) ? s.y : 0.0f;
        o.z = (m.z == c.z && s.z > 0.05f) ? s.z : 0.0f;
        o.w = (m.w == c.w && s.w > 0.05f) ? s.w : 0.0f;

        const int oy = row0 + rsel * 8 + k;
        float* op = out + planeBase + (long long)oy * W + x4;
        __builtin_nontemporal_store(o, (float4v*)op);

        h0 = h1;
        h1 = h2;
    }
}

extern "C" void kernel_launch(void* const* d_in, const int* in_sizes, int n_in,
                              void* d_out, int out_size, void* d_ws, size_t ws_size,
                              hipStream_t stream) {
    const float* heat = (const float*)d_in[0];
    // d_in[1] is kernel_size == 3 (hardcoded in the kernel: 3x3 window).
    float* out = (float*)d_out;

    const int planes = in_sizes[0] / (H * W);          // 16*17 = 272
    const int blocks = planes * (H / TILE_ROWS);       // 272 * 32 = 8704
    heatmap_peak_kernel<<<dim3(blocks), dim3(256), 0, stream>>>(heat, out);
}